// TaskAlignedAssigner_67972152426617
// MI455X (gfx1250) — compile-verified
//
#include <hip/hip_runtime.h>
#include <hip/hip_bf16.h>
#include <stdint.h>

// TaskAlignedAssigner for MI455X (gfx1250).
// B=32, M=40 gts, A=8400 anchors, C=80 classes. Memory-bound (~250MB traffic).
// WMMA f32 16x16x4 implements scores = pred_scores x onehot(cls) per batch.
// TDM (tensor_load_to_lds, 6-arg toolchain form) stages gt boxes into LDS.

#define NCLS 80
#define TOPK_ 10
#define BB 32
#define MM 40
#define AA 8400
#define EPS_F 1e-8f
#define IOU_EPS_F 1e-7f

typedef float v2f __attribute__((ext_vector_type(2)));
typedef float v8f __attribute__((ext_vector_type(8)));
typedef unsigned int u32x4 __attribute__((ext_vector_type(4)));
typedef int i32x4 __attribute__((ext_vector_type(4)));
typedef int i32x8 __attribute__((ext_vector_type(8)));

// ---------------------------------------------------------------------------
// Kernel 0: zero the pos_align / pos_iou stats (atomically maxed later).
// ---------------------------------------------------------------------------
__global__ void tal_init_pos(float* posA, float* posI) {
  int i = blockIdx.x * 256 + threadIdx.x;
  if (i < BB * MM) { posA[i] = 0.f; posI[i] = 0.f; }
}

// ---------------------------------------------------------------------------
// Kernel 1: scoresT[b,m,a] = pred_scores[b,a,cls[b,m]] via WMMA f32 16x16x4.
// D(16m x 16a) = sum_k A(16m x 4k onehot) * B(4k x 16a pred_scores^T).
// One wave per 16-anchor tile; 3 m-tiles (48 rows, top 8 dropped); 20 k-chunks.
// ---------------------------------------------------------------------------
__global__ __launch_bounds__(128) void tal_scores_wmma(
    const float* __restrict__ ps, const int* __restrict__ cls,
    float* __restrict__ scoresT) {
  const int wave  = threadIdx.x >> 5;
  const int lane  = threadIdx.x & 31;
  const int atile = blockIdx.x * 4 + wave;
  const int b     = blockIdx.y;
  if (atile >= AA / 16) return;  // 525 tiles

  const int col  = lane & 15;          // anchor column within tile / m row
  const int half = lane >> 4;          // 0: lanes 0-15, 1: lanes 16-31
  const int koff = half * 2;           // A/B frag: VGPR j holds K = j + koff
  const int aCol = atile * 16 + col;

  int cm0, cm1, cm2;                   // class ids for this lane's 3 m rows
  cm0 = cls[b * MM + col];             // rows 0..15
  cm1 = cls[b * MM + 16 + col];        // rows 16..31
  cm2 = (32 + col < MM) ? cls[b * MM + 32 + col] : -1;  // rows 32..39

  v8f acc0 = {}, acc1 = {}, acc2 = {};
  const float* psRow = ps + ((size_t)b * AA + aCol) * NCLS;

  for (int kc = 0; kc < NCLS / 4; ++kc) {
    const int c0 = kc * 4;
    const float2 pv = *reinterpret_cast<const float2*>(psRow + c0 + koff);
    v2f bf; bf.x = pv.x; bf.y = pv.y;
    const int ca = c0 + koff, cb = c0 + koff + 1;
    v2f a0, a1, a2;
    a0.x = (cm0 == ca) ? 1.f : 0.f;  a0.y = (cm0 == cb) ? 1.f : 0.f;
    a1.x = (cm1 == ca) ? 1.f : 0.f;  a1.y = (cm1 == cb) ? 1.f : 0.f;
    a2.x = (cm2 == ca) ? 1.f : 0.f;  a2.y = (cm2 == cb) ? 1.f : 0.f;
    acc0 = __builtin_amdgcn_wmma_f32_16x16x4_f32(false, a0, false, bf,
                                                 (short)0, acc0, false, false);
    acc1 = __builtin_amdgcn_wmma_f32_16x16x4_f32(false, a1, false, bf,
                                                 (short)0, acc1, false, false);
    acc2 = __builtin_amdgcn_wmma_f32_16x16x4_f32(false, a2, false, bf,
                                                 (short)0, acc2, false, false);
  }
  // D layout: VGPR i, lanes 0-15 -> row i, lanes 16-31 -> row i+8; col = lane&15
#pragma unroll
  for (int i = 0; i < 8; ++i) {
    const int r = i + 8 * half;
    scoresT[((size_t)b * MM + r) * AA + aCol]      = acc0[i];
    scoresT[((size_t)b * MM + 16 + r) * AA + aCol] = acc1[i];
    const int m2 = 32 + r;
    if (m2 < MM) scoresT[((size_t)b * MM + m2) * AA + aCol] = acc2[i];
  }
}

// ---------------------------------------------------------------------------
// Kernel 2: per (b,m) workgroup. Compute align = score*iou^6*valid into LDS
// (+ global, overwriting scoresT in place), run 10x block-argmax top-k
// (lowest-index tie break == jax.lax.top_k), write mask bytes = sel & in_gt.
// ---------------------------------------------------------------------------
__global__ __launch_bounds__(256) void tal_align_topk(
    const float* __restrict__ pbx, const float* __restrict__ ap,
    const float* __restrict__ gbx, const unsigned char* __restrict__ gmask,
    float* __restrict__ alignBuf, unsigned char* __restrict__ maskBuf) {
  const int m = blockIdx.x, b = blockIdx.y;
  const size_t base = ((size_t)b * MM + m) * (size_t)AA;
  if (!gmask[b * MM + m]) {   // invalid gt: mask row is all zero
    for (int a = threadIdx.x; a < AA; a += 256) maskBuf[base + a] = 0;
    return;
  }
  __shared__ float         alignLds[AA];
  __shared__ unsigned char ingtLds[AA];
  __shared__ unsigned char selLds[AA];
  __shared__ float redV[256];
  __shared__ int   redI[256];

  const float4 g = *reinterpret_cast<const float4*>(gbx + (b * MM + m) * 4);
  const float areaG = (g.z - g.x) * (g.w - g.y);

  for (int a = threadIdx.x; a < AA; a += 256) {
    const float2 p2 = *reinterpret_cast<const float2*>(ap + a * 2);
    const float dmin = fminf(fminf(p2.x - g.x, p2.y - g.y),
                             fminf(g.z - p2.x, g.w - p2.y));
    const bool ingt = dmin > EPS_F;
    const float4 p = *reinterpret_cast<const float4*>(pbx + ((size_t)b * AA + a) * 4);
    float iw = fmaxf(fminf(g.z, p.z) - fmaxf(g.x, p.x), 0.f);
    float ih = fmaxf(fminf(g.w, p.w) - fmaxf(g.y, p.y), 0.f);
    const float inter = iw * ih;
    const float areaP = (p.z - p.x) * (p.w - p.y);
    float iou = fmaxf(inter / (areaG + areaP - inter + IOU_EPS_F), 0.f);
    const float s = alignBuf[base + a];
    const float i2 = iou * iou;
    const float al = ingt ? (s * (i2 * i2 * i2)) : 0.f;
    alignBuf[base + a] = al;
    alignLds[a] = al;
    ingtLds[a]  = ingt ? 1 : 0;
    selLds[a]   = 0;
  }
  __syncthreads();

  for (int k = 0; k < TOPK_; ++k) {
    float bv = -1.f; int bi = 0;
    for (int a = threadIdx.x; a < AA; a += 256) {
      const float v = alignLds[a];
      if (v > bv) { bv = v; bi = a; }   // ascending scan -> lowest idx on tie
    }
    redV[threadIdx.x] = bv; redI[threadIdx.x] = bi;
    __syncthreads();
    for (int off = 128; off > 0; off >>= 1) {
      if (threadIdx.x < off) {
        const float ov = redV[threadIdx.x + off];
        const int   oi = redI[threadIdx.x + off];
        if (ov > redV[threadIdx.x] ||
            (ov == redV[threadIdx.x] && oi < redI[threadIdx.x])) {
          redV[threadIdx.x] = ov; redI[threadIdx.x] = oi;
        }
      }
      __syncthreads();
    }
    if (threadIdx.x == 0) {
      const int i = redI[0];
      selLds[i] = 1;
      alignLds[i] = -1.f;   // remove from further rounds (align >= 0 always)
    }
    __syncthreads();
  }
  for (int a = threadIdx.x; a < AA; a += 256)
    maskBuf[base + a] = (unsigned char)(selLds[a] & ingtLds[a]);
}

// ---------------------------------------------------------------------------
// IoU helper (matches kernel 2 formula; valid = in_gt & gtm folded in).
// ---------------------------------------------------------------------------
__device__ __forceinline__ float tal_iou(const float4& g, const float4& p,
                                         float ax, float ay, float gmv) {
  const float dmin = fminf(fminf(ax - g.x, ay - g.y), fminf(g.z - ax, g.w - ay));
  if (!(dmin > EPS_F) || !(gmv > 0.f)) return 0.f;
  const float iw = fmaxf(fminf(g.z, p.z) - fmaxf(g.x, p.x), 0.f);
  const float ih = fmaxf(fminf(g.w, p.w) - fmaxf(g.y, p.y), 0.f);
  const float inter = iw * ih;
  const float areaG = (g.z - g.x) * (g.w - g.y);
  const float areaP = (p.z - p.x) * (p.w - p.y);
  return fmaxf(inter / (areaG + areaP - inter + IOU_EPS_F), 0.f);
}

// ---------------------------------------------------------------------------
// Kernel 3: per (b,a) thread. gt boxes staged via Tensor Data Mover into LDS.
// Resolve multi-assignment (fg>1 -> argmax-iou one-hot), emit labels / boxes /
// fg, write masked align back in place, atomic-max pos_align / pos_iou.
// ---------------------------------------------------------------------------
__global__ __launch_bounds__(256) void tal_resolve(
    const float* __restrict__ pbx, const float* __restrict__ ap,
    const float* __restrict__ gbx, const int* __restrict__ cls,
    const unsigned char* __restrict__ gmask,
    float* __restrict__ alignBuf, unsigned char* __restrict__ maskBuf,
    float* __restrict__ posA, float* __restrict__ posI,
    float* __restrict__ outL, float* __restrict__ outB,
    float* __restrict__ outFg) {
  __shared__ __align__(16) float gbox[MM * 4];
  __shared__ int   clsL[MM];
  __shared__ float gmL[MM];
  const int b = blockIdx.y;

  if (threadIdx.x < 32) {
    // Tensor DMA descriptor: 1 tile of 160 x 1 f32 (gt_boxes[b]) -> LDS.
    const uint32_t lds = (uint32_t)(uintptr_t)&gbox[0];
    const uint64_t ga  = (uint64_t)(uintptr_t)(gbx + (size_t)b * MM * 4);
    u32x4 g0; i32x8 g1; i32x4 g2, g3; i32x8 g4;
    g0[0] = 1u;                                     // count=1 (valid), no gather
    g0[1] = lds;                                    // lds_addr [63:32]
    g0[2] = (uint32_t)(ga & 0xffffffffu);           // global_addr lo
    g0[3] = (uint32_t)((ga >> 32) & 0x01ffffffu)    // global_addr hi
            | (2u << 30);                           // type=2 (image)
    g1[0] = (int)(2u << 16);                        // data_size=4B; wg_mask=0
    g1[1] = (int)(160u << 16);                      // tensor_dim0[15:0]
    g1[2] = (int)(1u << 16);                        // dim0 hi=0 | tensor_dim1 lo=1
    g1[3] = (int)(160u << 16);                      // dim1 hi=0 | tile_dim0=160
    g1[4] = 1;                                      // tile_dim1=1, tile_dim2=0
    g1[5] = 160;                                    // tensor_dim0_stride lo
    g1[6] = 0; g1[7] = 0;                           // stride hi, dim1_stride
    g2[0] = g2[1] = g2[2] = g2[3] = 0;              // group2 (2D tensor: unused)
    g3[0] = g3[1] = g3[2] = g3[3] = 0;              // group3 (unused)
    g4[0] = g4[1] = g4[2] = g4[3] = 0;              // extra group (6-arg form)
    g4[4] = g4[5] = g4[6] = g4[7] = 0;
    __builtin_amdgcn_tensor_load_to_lds(g0, g1, g2, g3, g4, 0);
    __builtin_amdgcn_s_wait_tensorcnt(0);
  }
  if (threadIdx.x >= 32 && threadIdx.x < 32 + MM) {
    const int m = threadIdx.x - 32;
    clsL[m] = cls[b * MM + m];
    gmL[m]  = gmask[b * MM + m] ? 1.f : 0.f;
  }
  __syncthreads();

  const int a = blockIdx.x * 256 + threadIdx.x;
  if (a >= AA) return;

  const float2 p2 = *reinterpret_cast<const float2*>(ap + a * 2);
  const float4 p  = *reinterpret_cast<const float4*>(pbx + ((size_t)b * AA + a) * 4);

  uint64_t bits = 0; int fg0 = 0;
  float bestIou = -1.f; int bestM = 0;
  for (int m = 0; m < MM; ++m) {
    const unsigned char mk = maskBuf[((size_t)b * MM + m) * AA + a];
    const float4 g = *reinterpret_cast<const float4*>(&gbox[m * 4]);
    const float iou = tal_iou(g, p, p2.x, p2.y, gmL[m]);
    if (iou > bestIou) { bestIou = iou; bestM = m; }   // first-max == argmax
    if (mk) { bits |= (1ull << m); ++fg0; }
  }
  int fgF = fg0;
  if (fg0 > 1) { bits = (1ull << bestM); fgF = 1; }    // max-iou one-hot

  const int tgt = bits ? (__ffsll((unsigned long long)bits) - 1) : 0;
  int lbl = clsL[tgt]; lbl = lbl < 0 ? 0 : (lbl > NCLS ? NCLS : lbl);
  outL[(size_t)b * AA + a]  = (float)lbl;
  const float4 tb = *reinterpret_cast<const float4*>(&gbox[tgt * 4]);
  *reinterpret_cast<float4*>(outB + ((size_t)b * AA + a) * 4) = tb;
  outFg[(size_t)b * AA + a] = (fgF > 0) ? 1.f : 0.f;

  for (int m = 0; m < MM; ++m) {
    const size_t idx = ((size_t)b * MM + m) * AA + a;
    const unsigned char bit = (unsigned char)((bits >> m) & 1ull);
    const float alv = alignBuf[idx];
    const float w = bit ? alv : 0.f;
    alignBuf[idx] = w;         // align * final_mask
    maskBuf[idx]  = bit;
    if (bit) {
      const float4 g = *reinterpret_cast<const float4*>(&gbox[m * 4]);
      const float iou = tal_iou(g, p, p2.x, p2.y, gmL[m]);
      // values >= 0: int-punned atomicMax preserves float ordering
      atomicMax((int*)&posA[b * MM + m], __float_as_int(w));
      atomicMax((int*)&posI[b * MM + m], __float_as_int(iou));
    }
  }
}

// ---------------------------------------------------------------------------
// Kernel 4: norm = max_m(align_masked * pos_iou / (pos_align + eps));
// target_scores[b,a,:] = onehot(label) * norm * (fg>0), float4-streamed.
// ---------------------------------------------------------------------------
__global__ __launch_bounds__(256) void tal_scores_out(
    const float* __restrict__ alignBuf, const float* __restrict__ posA,
    const float* __restrict__ posI, const float* __restrict__ outL,
    const float* __restrict__ outFg, float* __restrict__ outS) {
  __shared__ float pA[MM], pI[MM];
  const int b = blockIdx.y;
  if (threadIdx.x < MM) {
    pA[threadIdx.x] = posA[b * MM + threadIdx.x];
    pI[threadIdx.x] = posI[b * MM + threadIdx.x];
  }
  __syncthreads();
  const int a = blockIdx.x * 256 + threadIdx.x;
  if (a >= AA) return;

  float norm = 0.f;
  for (int m = 0; m < MM; ++m) {
    const float t = alignBuf[((size_t)b * MM + m) * AA + a] * pI[m] / (pA[m] + EPS_F);
    norm = fmaxf(norm, t);
  }
  const int lbl = (int)outL[(size_t)b * AA + a];
  const float val = (outFg[(size_t)b * AA + a] > 0.f) ? norm : 0.f;
  float* row = outS + ((size_t)b * AA + a) * NCLS;
#pragma unroll
  for (int c0 = 0; c0 < NCLS; c0 += 4) {
    float4 v = {0.f, 0.f, 0.f, 0.f};
    if (lbl >= c0 && lbl < c0 + 4) (&v.x)[lbl - c0] = val;
    *reinterpret_cast<float4*>(row + c0) = v;
  }
}

// ---------------------------------------------------------------------------
extern "C" void kernel_launch(void* const* d_in, const int* in_sizes, int n_in,
                              void* d_out, int out_size, void* d_ws,
                              size_t ws_size, hipStream_t stream) {
  const float*         ps = (const float*)d_in[0];          // (B,A,C)
  const float*         pb = (const float*)d_in[1];          // (B,A,4)
  const float*         ap = (const float*)d_in[2];          // (A,2)
  const int*           gl = (const int*)d_in[3];            // (B,M,1)
  const float*         gb = (const float*)d_in[4];          // (B,M,4)
  const unsigned char* gm = (const unsigned char*)d_in[5];  // (B,M,1) bool

  char* ws = (char*)d_ws;
  const size_t alignBytes = (size_t)BB * MM * AA * sizeof(float);  // 43.0 MB
  const size_t maskBytes  = (size_t)BB * MM * AA;                  // 10.8 MB
  float*         alignBuf = (float*)ws;                       // scoresT -> align
  unsigned char* maskBuf  = (unsigned char*)(ws + alignBytes);
  float*         posA     = (float*)(ws + alignBytes + maskBytes);
  float*         posI     = posA + BB * MM;

  float* out   = (float*)d_out;
  float* outL  = out;                          // (B,A)   labels (as float)
  float* outB  = out + (size_t)BB * AA;        // (B,A,4) boxes
  float* outS  = out + (size_t)BB * AA * 5;    // (B,A,C) scores
  float* outFg = out + (size_t)BB * AA * 85;   // (B,A)   fg flag

  tal_init_pos<<<dim3((BB * MM + 255) / 256), 256, 0, stream>>>(posA, posI);
  tal_scores_wmma<<<dim3((AA / 16 + 3) / 4, BB), 128, 0, stream>>>(ps, gl, alignBuf);
  tal_align_topk<<<dim3(MM, BB), 256, 0, stream>>>(pb, ap, gb, gm, alignBuf, maskBuf);
  tal_resolve<<<dim3((AA + 255) / 256, BB), 256, 0, stream>>>(
      pb, ap, gb, gl, gm, alignBuf, maskBuf, posA, posI, outL, outB, outFg);
  tal_scores_out<<<dim3((AA + 255) / 256, BB), 256, 0, stream>>>(
      alignBuf, posA, posI, outL, outFg, outS);
}